// GyroBNH_50766513438860
// MI455X (gfx1250) — compile-verified
//
#include <hip/hip_runtime.h>
#include <math.h>

#define EPSF  1e-6f
#define MINN  1e-15f
#define MAXN  (1.0f - 1e-5f)
#define D     32
#define TILE_PTS 256
#define ROW   33              // padded LDS row stride (conflict-free dots)
#define NTHREADS 256

typedef __attribute__((ext_vector_type(2))) float v2f;
typedef __attribute__((ext_vector_type(8))) float v8f;

__device__ __forceinline__ float atanh_f(float z) {
    return 0.5f * __logf((1.0f + z) / (1.0f - z));
}
__device__ __forceinline__ float tanh_f(float z) {
    // 1 - 2/(e^{2z}+1): safe at +/-inf of exp
    return 1.0f - 2.0f / (__expf(2.0f * z) + 1.0f);
}
__device__ __forceinline__ float clampf(float v, float lo, float hi) {
    return fminf(fmaxf(v, lo), hi);
}

// ---------------------------------------------------------------------------
// Streaming pass kernel.
// MODE 0: colsum over N of  atanh(|x|)/|x| * x            (log0 mean init)
// MODE 1: colsum over N of  cB*x  and scalar sum of cA,   where per point
//         logmap(mu,x) = cA*mu + cB*x   (Frechet iteration body)
// MODE 2: scalar sum over N of (2*atanh(|mobius_add(-mu,x)|))^2  (variance)
// Column sums are computed with V_WMMA_F32_16X16X4_F32, A = all-ones, so the
// 16x16 fp32 accumulator carries the exact fp32 column reduction.
// ---------------------------------------------------------------------------
template <int MODE>
__global__ void pass_kernel(const float* __restrict__ x,
                            const float* __restrict__ muG,
                            float* __restrict__ partV,
                            float* __restrict__ partA,
                            int N, int numTiles) {
    __shared__ float tile[TILE_PTS * ROW];
    __shared__ float sArr[TILE_PTS];
    __shared__ float waveAcc[8][D];
    __shared__ float red[NTHREADS];
    __shared__ float muSh[D];

    const int tid  = threadIdx.x;
    const int lane = tid & 31;
    const int wave = tid >> 5;

    float m2 = 0.0f;
    if (MODE >= 1) {
        if (tid < D) muSh[tid] = muG[tid];
        __syncthreads();
        #pragma unroll
        for (int d = 0; d < D; ++d) m2 += muSh[d] * muSh[d];
    }

    v8f cLo = {}; v8f cHi = {};
    float aAcc = 0.0f;
    v2f ones; ones.x = 1.0f; ones.y = 1.0f;

    const long long totElem = (long long)N * D;

    for (int t = blockIdx.x; t < numTiles; t += gridDim.x) {
        const long long base4 = (long long)t * (TILE_PTS * D / 4);
        // prefetch next tile (global_prefetch_b8 path)
        if (t + (int)gridDim.x < numTiles) {
            const float* nxt = x + ((long long)(t + gridDim.x) * TILE_PTS * D) + tid * 4;
            __builtin_prefetch(nxt, 0, 1);
        }
        // coalesced b128 load -> padded LDS tile
        #pragma unroll
        for (int i = 0; i < 8; ++i) {
            const long long e4 = (long long)tid + (long long)i * NTHREADS; // 0..2047
            const long long ge = base4 + e4;
            const int p  = (int)(e4 >> 3);
            const int dd = (int)((e4 & 7) << 2);
            float4 v;
            if (ge * 4 + 3 < totElem) v = ((const float4*)x)[ge];
            else { v.x = 0.f; v.y = 0.f; v.z = 0.f; v.w = 0.f; }
            tile[p * ROW + dd + 0] = v.x;
            tile[p * ROW + dd + 1] = v.y;
            tile[p * ROW + dd + 2] = v.z;
            tile[p * ROW + dd + 3] = v.w;
        }
        __syncthreads();

        // per-point scalar coefficients (thread tid <-> point tid of the tile)
        const int gp = t * TILE_PTS + tid;
        float s = 0.0f;
        if (MODE == 0) {
            float x2 = 0.0f;
            #pragma unroll
            for (int d = 0; d < D; ++d) { float xv = tile[tid * ROW + d]; x2 += xv * xv; }
            float n = clampf(sqrtf(x2), MINN, MAXN);
            s = atanh_f(n) / n;
            if (gp >= N) s = 0.0f;
        } else {
            float x2 = 0.0f, a = 0.0f;
            #pragma unroll
            for (int d = 0; d < D; ++d) {
                float xv = tile[tid * ROW + d];
                x2 += xv * xv; a += muSh[d] * xv;
            }
            const float alpha = 1.0f - 2.0f * a + x2;
            const float beta  = 1.0f - m2;
            const float den   = fmaxf(1.0f - 2.0f * a + m2 * x2, MINN);
            const float nn    = fmaxf(alpha * alpha * m2 - 2.0f * alpha * beta * a
                                      + beta * beta * x2, 0.0f);
            const float wn    = sqrtf(nn) / den;
            if (MODE == 1) {
                const float nc = clampf(wn, MINN, MAXN);
                const float sc = beta * atanh_f(nc) / nc;   // (1-m2)*atanh(n)/n
                float cB = sc * beta / den;
                float cA = -sc * alpha / den;
                if (gp >= N) { cB = 0.0f; cA = 0.0f; }
                s = cB;
                aAcc += cA;
            } else { // MODE 2: variance
                const float nc = clampf(wn, 0.0f, MAXN);
                const float dg = 2.0f * atanh_f(nc);
                if (gp < N) aAcc += dg * dg;
            }
        }
        sArr[tid] = s;
        __syncthreads();

        if (MODE <= 1) {
            // WMMA column-sum: this wave reduces its own 32 points
            const int half = lane >> 4;
            const int n16  = lane & 15;
            const int pb   = wave * 32;
            #pragma unroll
            for (int g = 0; g < 8; ++g) {
                const int p0 = pb + g * 4 + half * 2;
                const int p1 = p0 + 1;
                const float s0 = sArr[p0], s1 = sArr[p1];
                v2f bLo, bHi;
                bLo.x = s0 * tile[p0 * ROW + n16];
                bLo.y = s1 * tile[p1 * ROW + n16];
                bHi.x = s0 * tile[p0 * ROW + 16 + n16];
                bHi.y = s1 * tile[p1 * ROW + 16 + n16];
                cLo = __builtin_amdgcn_wmma_f32_16x16x4_f32(
                        false, ones, false, bLo, (short)0, cLo, false, false);
                cHi = __builtin_amdgcn_wmma_f32_16x16x4_f32(
                        false, ones, false, bHi, (short)0, cHi, false, false);
            }
        }
        __syncthreads();
    }

    // combine waves -> per-block partials
    if (MODE <= 1) {
        if (lane < 16) {                    // row M=0 of C: lanes 0-15, VGPR 0
            waveAcc[wave][lane]      = cLo[0];
            waveAcc[wave][16 + lane] = cHi[0];
        }
    }
    red[tid] = aAcc;
    __syncthreads();
    if (MODE <= 1 && tid < D) {
        float sum = 0.0f;
        #pragma unroll
        for (int w = 0; w < 8; ++w) sum += waveAcc[w][tid];
        partV[blockIdx.x * D + tid] = sum;
    }
    if (MODE >= 1) {
        for (int stride = NTHREADS / 2; stride > 0; stride >>= 1) {
            __syncthreads();
            if (tid < stride) red[tid] += red[tid + stride];
        }
        if (tid == 0) partA[blockIdx.x] = red[0];
    }
}

// ---------------------------------------------------------------------------
// Finalize kernels: single wave, 32 threads (thread d <-> dim d)
// ---------------------------------------------------------------------------
__global__ void finalize_init(const float* __restrict__ partV,
                              float* __restrict__ muG, int N, int G) {
    __shared__ float sh[D];
    const int d = threadIdx.x;
    float s = 0.0f;
    for (int b = 0; b < G; ++b) s += partV[b * D + d];
    const float v = s / (float)N;
    sh[d] = v * v;
    __syncthreads();
    float n2 = 0.0f;
    for (int i = 0; i < D; ++i) n2 += sh[i];
    const float n = fmaxf(sqrtf(n2), MINN);
    muG[d] = tanh_f(n) * v / n;              // exp0(v)
}

__global__ void finalize_iter(const float* __restrict__ partV,
                              const float* __restrict__ partA,
                              float* __restrict__ muG, int N, int G) {
    __shared__ float mu[D], v[D], bc[1];
    const int d = threadIdx.x;
    mu[d] = muG[d];
    float sv = 0.0f;
    for (int b = 0; b < G; ++b) sv += partV[b * D + d];
    if (d == 0) {
        float sa = 0.0f;
        for (int b = 0; b < G; ++b) sa += partA[b];
        bc[0] = sa;
    }
    __syncthreads();
    const float sa = bc[0];
    const float vd = (sa * mu[d] + sv) / (float)N;   // mean of logmap
    v[d] = vd;
    __syncthreads();
    float nv2 = 0.0f, m2 = 0.0f, muv = 0.0f;
    for (int i = 0; i < D; ++i) {
        nv2 += v[i] * v[i]; m2 += mu[i] * mu[i]; muv += mu[i] * v[i];
    }
    const float nv = fmaxf(sqrtf(nv2), MINN);
    const float th = tanh_f(nv / (1.0f - m2));
    const float kt = th / nv;
    const float ud = kt * vd;
    const float xy = kt * muv;               // <mu,u>
    const float u2 = kt * kt * nv2;          // |u|^2
    const float num = (1.0f + 2.0f * xy + u2) * mu[d] + (1.0f - m2) * ud;
    const float den = fmaxf(1.0f + 2.0f * xy + m2 * u2, MINN);
    __syncthreads();
    muG[d] = num / den;                      // exp_map(mu, v)
}

__global__ void finalize_var(const float* __restrict__ partA,
                             const float* __restrict__ mp,
                             const float* __restrict__ vp,
                             const float* __restrict__ muG,
                             float* __restrict__ mG,
                             float* __restrict__ scal,
                             float* __restrict__ outTail,
                             int N, int G) {
    __shared__ float sh[D];
    const int d = threadIdx.x;
    sh[d] = mp[d] * mp[d];
    __syncthreads();
    float n2 = 0.0f;
    for (int i = 0; i < D; ++i) n2 += sh[i];
    const float n  = fmaxf(sqrtf(n2), MINN);
    const float md = tanh_f(n) * mp[d] / n;  // on_manifold = exp0(mean_param)
    mG[d] = md;
    sh[d] = md * md;
    __syncthreads();
    float mm2 = 0.0f;
    for (int i = 0; i < D; ++i) mm2 += sh[i];
    __syncthreads();
    sh[d] = md * muG[d];
    __syncthreads();
    float mmu = 0.0f;
    for (int i = 0; i < D; ++i) mmu += sh[i];
    if (d == 0) {
        float sv = 0.0f;
        for (int b = 0; b < G; ++b) sv += partA[b];
        const float var = sv / (float)N;
        scal[0] = vp[0] / sqrtf(var + EPSF);   // factor
        scal[1] = mm2;
        scal[2] = mmu;
    }
    outTail[d] = muG[d];                        // second tuple output: input_mean
}

// ---------------------------------------------------------------------------
// Final elementwise pass: out = mobius_add(m, mobius_scalar_mul(f, mobius_add(-mu,x)))
// expressed per point as out[d] = P*m[d] + Q*mu[d] + R*x[d]
// ---------------------------------------------------------------------------
__global__ void normalize_kernel(const float* __restrict__ x,
                                 const float* __restrict__ muG,
                                 const float* __restrict__ mG,
                                 const float* __restrict__ scal,
                                 float* __restrict__ out,
                                 int N, int numTiles) {
    __shared__ float tile[TILE_PTS * ROW];
    __shared__ float muSh[D], mSh[D];
    const int tid = threadIdx.x;
    if (tid < D) { muSh[tid] = muG[tid]; mSh[tid] = mG[tid]; }
    __syncthreads();
    const float factor = scal[0], mm2 = scal[1], mmu = scal[2];
    float m2 = 0.0f;
    #pragma unroll
    for (int d = 0; d < D; ++d) m2 += muSh[d] * muSh[d];

    const long long totElem = (long long)N * D;

    for (int t = blockIdx.x; t < numTiles; t += gridDim.x) {
        const long long base4 = (long long)t * (TILE_PTS * D / 4);
        #pragma unroll
        for (int i = 0; i < 8; ++i) {
            const long long e4 = (long long)tid + (long long)i * NTHREADS;
            const long long ge = base4 + e4;
            const int p  = (int)(e4 >> 3);
            const int dd = (int)((e4 & 7) << 2);
            float4 v;
            if (ge * 4 + 3 < totElem) v = ((const float4*)x)[ge];
            else { v.x = 0.f; v.y = 0.f; v.z = 0.f; v.w = 0.f; }
            tile[p * ROW + dd + 0] = v.x;
            tile[p * ROW + dd + 1] = v.y;
            tile[p * ROW + dd + 2] = v.z;
            tile[p * ROW + dd + 3] = v.w;
        }
        __syncthreads();

        float x2 = 0.0f, a = 0.0f, hm = 0.0f;
        #pragma unroll
        for (int d = 0; d < D; ++d) {
            const float xv = tile[tid * ROW + d];
            x2 += xv * xv; a += muSh[d] * xv; hm += mSh[d] * xv;
        }
        const float alpha = 1.0f - 2.0f * a + x2;
        const float beta  = 1.0f - m2;
        const float den   = fmaxf(1.0f - 2.0f * a + m2 * x2, MINN);
        const float nn    = fmaxf(alpha * alpha * m2 - 2.0f * alpha * beta * a
                                  + beta * beta * x2, 0.0f);
        const float wn    = sqrtf(nn) / den;
        const float nc    = clampf(wn, MINN, MAXN);
        const float tq    = tanh_f(factor * atanh_f(nc));
        const float ky    = tq / nc;
        const float kA    = -ky * alpha / den;
        const float kB    =  ky * beta  / den;
        const float y2    = ky * ky * nn / (den * den);     // |y|^2
        const float b     = kA * mmu + kB * hm;             // <m,y>
        const float ddn   = fmaxf(1.0f + 2.0f * b + mm2 * y2, MINN);
        const float P     = (1.0f + 2.0f * b + y2) / ddn;
        const float Q     = (1.0f - mm2) * kA / ddn;
        const float R     = (1.0f - mm2) * kB / ddn;
        #pragma unroll
        for (int d = 0; d < D; ++d) {
            tile[tid * ROW + d] = P * mSh[d] + Q * muSh[d] + R * tile[tid * ROW + d];
        }
        __syncthreads();

        #pragma unroll
        for (int i = 0; i < 8; ++i) {
            const long long e4 = (long long)tid + (long long)i * NTHREADS;
            const long long ge = base4 + e4;
            if (ge * 4 + 3 < totElem) {
                const int p  = (int)(e4 >> 3);
                const int dd = (int)((e4 & 7) << 2);
                float4 v;
                v.x = tile[p * ROW + dd + 0];
                v.y = tile[p * ROW + dd + 1];
                v.z = tile[p * ROW + dd + 2];
                v.w = tile[p * ROW + dd + 3];
                ((float4*)out)[ge] = v;
            }
        }
        __syncthreads();
    }
}

// ---------------------------------------------------------------------------
extern "C" void kernel_launch(void* const* d_in, const int* in_sizes, int n_in,
                              void* d_out, int out_size, void* d_ws, size_t ws_size,
                              hipStream_t stream) {
    const float* x  = (const float*)d_in[0];
    const float* mp = (const float*)d_in[1];
    const float* vp = (const float*)d_in[2];
    float* out = (float*)d_out;

    const int N = in_sizes[0] / D;
    const int numTiles = (N + TILE_PTS - 1) / TILE_PTS;

    // workspace layout (floats): [0..31] mu  [32..63] m  [64..71] scalars
    // [128 .. 128+G*32) partV   [128+G*32 .. +G) partA
    int G = 512;
    {
        const size_t need = (size_t)(128 + G * D + G) * sizeof(float);
        while (G > 32 && need > ws_size) G >>= 1;   // defensive clamp
    }
    float* ws    = (float*)d_ws;
    float* muG   = ws;
    float* mG    = ws + 32;
    float* scal  = ws + 64;
    float* partV = ws + 128;
    float* partA = ws + 128 + (size_t)G * D;

    dim3 grid(G), block(NTHREADS);

    // pass 1: tangent-space mean init, mu0 = exp0(mean(log0(x)))
    pass_kernel<0><<<grid, block, 0, stream>>>(x, muG, partV, partA, N, numTiles);
    finalize_init<<<1, D, 0, stream>>>(partV, muG, N, G);

    // 10 Frechet iterations
    for (int it = 0; it < 10; ++it) {
        pass_kernel<1><<<grid, block, 0, stream>>>(x, muG, partV, partA, N, numTiles);
        finalize_iter<<<1, D, 0, stream>>>(partV, partA, muG, N, G);
    }

    // variance + scalars; also writes input_mean into d_out tail
    pass_kernel<2><<<grid, block, 0, stream>>>(x, muG, partV, partA, N, numTiles);
    finalize_var<<<1, D, 0, stream>>>(partA, mp, vp, muG, mG, scal,
                                      out + (size_t)N * D, N, G);

    // final normalize pass
    normalize_kernel<<<grid, block, 0, stream>>>(x, muG, mG, scal, out, N, numTiles);
}